// EdgePreservingRefinement_82446192214518
// MI455X (gfx1250) — compile-verified
//
#include <hip/hip_runtime.h>
#include <math.h>

// ---------------- geometry ----------------
#define PW     512
#define PLANE  (PW*PW)        // 262144
#define NPL    16             // B*C planes
#define TOT    (NPL*PLANE)    // 4194304 floats per full image stack

typedef float v2f __attribute__((ext_vector_type(2)));
typedef float v8f __attribute__((ext_vector_type(8)));
typedef int   v4i __attribute__((ext_vector_type(4)));

#if __has_builtin(__builtin_amdgcn_wmma_f32_16x16x4_f32)
#define USE_WMMA 1
#else
#define USE_WMMA 0
#endif

#if __has_builtin(__builtin_amdgcn_global_load_async_to_lds_b128)
#define HAVE_ASYNC 1
#else
#define HAVE_ASYNC 0
#endif

#if HAVE_ASYNC
__device__ __forceinline__ void async_g2l_b128(const float* g, float* l)
{
    // GLOBAL_LOAD_ASYNC_TO_LDS_B128: per-lane 16B global -> LDS, ASYNCcnt-tracked
    __builtin_amdgcn_global_load_async_to_lds_b128(
        (__attribute__((address_space(1))) v4i*)(g),
        (__attribute__((address_space(3))) v4i*)(l), 0, 0);
}
__device__ __forceinline__ void async_wait0()
{
#if __has_builtin(__builtin_amdgcn_s_wait_asynccnt)
    __builtin_amdgcn_s_wait_asynccnt(0);
#else
    asm volatile("s_wait_asynccnt 0" ::: "memory");
#endif
}
#endif

// Gaussian taps (sigma 0.5 / 1.0 / 2.0), normalized as in the reference (float32)
__constant__ float G5c[5]   = {2.63865e-4f, 0.10645079f, 0.78657073f, 0.10645079f, 2.63865e-4f};
__constant__ float G7c[7]   = {0.00443305f, 0.05400558f, 0.24203623f, 0.39905028f,
                               0.24203623f, 0.05400558f, 0.00443305f};
__constant__ float G13c[13] = {0.00221817f, 0.00877313f, 0.02702316f, 0.06482519f,
                               0.12110939f, 0.17621312f, 0.19967563f, 0.17621312f,
                               0.12110939f, 0.06482519f, 0.02702316f, 0.00877313f,
                               0.00221817f};

// ============================================================================
// K1: multiscale Laplacian edge map (3 gaussian blurs + lap3/lap5, softmax mix)
// ============================================================================
__global__ __launch_bounds__(256)
void k_edges(const float* __restrict__ x, const float* __restrict__ sw,
             float* __restrict__ eout)
{
    const int tx = blockIdx.x << 4, ty = blockIdx.y << 4, pl = blockIdx.z;
    const size_t pb = (size_t)pl * PLANE;
    const int tid = threadIdx.x, lx = tid & 15, ly = tid >> 4;

    __shared__ float sIn[32 * 33];
    __shared__ float sV [20 * 32];
    __shared__ float sBl[20 * 20];

    for (int i = tid; i < 32 * 32; i += 256) {
        int r = i >> 5, c = i & 31;
        int gr = ty - 8 + r, gc = tx - 8 + c;
        bool ok = ((unsigned)gr < (unsigned)PW) & ((unsigned)gc < (unsigned)PW);
        sIn[r * 33 + c] = ok ? x[pb + (size_t)gr * PW + gc] : 0.0f;
    }
    __syncthreads();

    float s0 = sw[0], s1 = sw[1], s2 = sw[2];
    float mx = fmaxf(s0, fmaxf(s1, s2));
    float e0 = expf(s0 - mx), e1 = expf(s1 - mx), e2 = expf(s2 - mx);
    float inv = 1.0f / (e0 + e1 + e2);
    const float wsc[3] = {e0 * inv, e1 * inv, e2 * inv};

    float fused = 0.0f;
    for (int sc = 0; sc < 3; ++sc) {
        const float* g = (sc == 0) ? G5c : (sc == 1) ? G7c : G13c;
        const int grd = (sc == 0) ? 2 : (sc == 1) ? 3 : 6;   // gauss radius
        const int lr  = (sc == 0) ? 1 : 2;                   // laplacian radius
        const int VR = 16 + 2 * lr;
        const int VC = 16 + 2 * (lr + grd);
        const int BC = 16 + 2 * lr;

        for (int i = tid; i < VR * VC; i += 256) {
            int vr = i / VC, vc = i - vr * VC;
            int rr = vr - lr, cc = vc - lr - grd;
            float s = 0.0f;
            for (int k = 0; k <= 2 * grd; ++k)
                s += g[k] * sIn[(rr - grd + k + 8) * 33 + (cc + 8)];
            sV[i] = s;
        }
        __syncthreads();
        for (int i = tid; i < BC * BC; i += 256) {
            int br = i / BC, bc = i - br * BC;
            float s = 0.0f;
            for (int k = 0; k <= 2 * grd; ++k)
                s += g[k] * sV[br * VC + bc + k];
            int g2r = ty + br - lr, g2c = tx + bc - lr;
            bool ok = ((unsigned)g2r < (unsigned)PW) & ((unsigned)g2c < (unsigned)PW);
            sBl[i] = ok ? s : 0.0f;   // match zero-padded laplacian input at borders
        }
        __syncthreads();

        int o = (ly + lr) * BC + (lx + lr);
        float l;
        if (lr == 1) {
            l = sBl[o - BC] + sBl[o + BC] + sBl[o - 1] + sBl[o + 1] - 4.0f * sBl[o];
        } else {
            l = -16.0f * sBl[o]
              + 2.0f * (sBl[o - 1] + sBl[o + 1] + sBl[o - BC] + sBl[o + BC])
              + (sBl[o - 2] + sBl[o + 2] + sBl[o - 2 * BC] + sBl[o + 2 * BC])
              + (sBl[o - BC - 1] + sBl[o - BC + 1] + sBl[o + BC - 1] + sBl[o + BC + 1]);
        }
        fused += wsc[sc] * fabsf(l);
        __syncthreads();
    }
    eout[pb + (size_t)(ty + ly) * PW + (tx + lx)] = fused;
}

// ============================================================================
// K2/K4: 31x31 box mean via V_WMMA_F32_16X16X4_F32 banded matmuls.
// One wave per 16x16 output tile; 48-row x 52-col halo tile in LDS
// (columns start at tx-16 so GLOBAL_LOAD_ASYNC_TO_LDS_B128 is 16B-aligned;
//  LDS row stride 56 floats keeps every row 16B-aligned).
// H-pass: D = sum_b Data_b(16x16) x Band_b   V-pass: D = sum_b Band_b x H_b
// Band = shifted ones/31 diagonals. mode 0: {I, p, I*p, I*I}; mode 1: {a, b}.
// ============================================================================
#define BOX_ST 56   // LDS row stride (floats)

__device__ __forceinline__ float chan_val(const float* sA, const float* sB,
                                          int mode, int ch, int idx)
{
    float a = sA[idx], b = sB[idx];
    if (mode == 0) {
        if (ch == 0) return a;
        if (ch == 1) return b;
        if (ch == 2) return a * b;
        return a * a;
    }
    return (ch == 0) ? a : b;
}

__global__ __launch_bounds__(32)
void k_box31(const float* __restrict__ in0, const float* __restrict__ in1,
             float* __restrict__ o0, float* __restrict__ o1,
             float* __restrict__ o2, float* __restrict__ o3,
             int nch, int mode)
{
    const int tx = blockIdx.x << 4, ty = blockIdx.y << 4, pl = blockIdx.z;
    const size_t pb = (size_t)pl * PLANE;
    const int lane = threadIdx.x;
    const int half = lane >> 4, ln = lane & 15;
    const float inv31 = 1.0f / 31.0f;

    __shared__ float sA[48 * BOX_ST];
    __shared__ float sB[48 * BOX_ST];
    __shared__ float sH[48 * 17];

    // rows ty-15 .. ty+32 ; cols tx-16 .. tx+35 (52 wide, 16-float aligned start)
    const bool interior = (tx >= 16) && (tx <= PW - 48) && (ty >= 16) && (ty <= PW - 48);

#if HAVE_ASYNC
    if (interior) {
        // 13 vec4 per row x 48 rows = 624 B128 transfers per channel
        for (int i = lane; i < 48 * 13; i += 32) {
            int r = i / 13, c4 = (i - r * 13) << 2;
            size_t gi = pb + (size_t)(ty - 15 + r) * PW + (tx - 16 + c4);
            int li = r * BOX_ST + c4;
            async_g2l_b128(&in0[gi], &sA[li]);
            async_g2l_b128(&in1[gi], &sB[li]);
        }
        async_wait0();
    } else
#endif
    {
        __builtin_prefetch(&in0[pb + (size_t)ty * PW + tx], 0, 1);  // global_prefetch
        __builtin_prefetch(&in1[pb + (size_t)ty * PW + tx], 0, 1);
        for (int i = lane; i < 48 * 52; i += 32) {
            int r = i / 52, c = i - r * 52;
            int gr = ty - 15 + r, gc = tx - 16 + c;
            bool ok = ((unsigned)gr < (unsigned)PW) & ((unsigned)gc < (unsigned)PW);
            size_t gi = pb + (size_t)gr * PW + gc;
            sA[r * BOX_ST + c] = ok ? in0[gi] : 0.0f;
            sB[r * BOX_ST + c] = ok ? in1[gi] : 0.0f;
        }
    }
    __syncthreads();

    for (int ch = 0; ch < nch; ++ch) {
        // ---- horizontal pass: H[r][n], r = 0..47 halo rows, n = 0..15 tile cols
        for (int g = 0; g < 3; ++g) {
#if USE_WMMA
            v8f acc = {};
            #pragma unroll
            for (int b = 0; b < 3; ++b) {
                #pragma unroll
                for (int t = 0; t < 4; ++t) {
                    const int kk  = 4 * t + 2 * half;   // this lane's K pair
                    const int row = 16 * g + ln;        // A-matrix M = lane%16
                    const int cb  = 16 * b + kk;        // matrix column index
                    // local LDS col = cb + 1 (window starts one col earlier)
                    v2f av = { chan_val(sA, sB, mode, ch, row * BOX_ST + cb + 1),
                               chan_val(sA, sB, mode, ch, row * BOX_ST + cb + 2) };
                    // band: W[c][n] != 0 iff n <= 16b+c <= n+30
                    v2f bv = { (cb     >= ln && cb     <= ln + 30) ? inv31 : 0.0f,
                               (cb + 1 >= ln && cb + 1 <= ln + 30) ? inv31 : 0.0f };
                    acc = __builtin_amdgcn_wmma_f32_16x16x4_f32(
                              false, av, false, bv, (short)0, acc, false, false);
                }
            }
            #pragma unroll
            for (int v = 0; v < 8; ++v)          // C/D layout: row = v + 8*half
                sH[(16 * g + v + 8 * half) * 17 + ln] = acc[v];
#else
            for (int e = lane; e < 16 * 16; e += 32) {
                int r = 16 * g + (e >> 4), n = e & 15;
                float s = 0.0f;
                for (int j = 0; j < 31; ++j)
                    s += chan_val(sA, sB, mode, ch, r * BOX_ST + n + j + 1);
                sH[r * 17 + n] = s * inv31;
            }
#endif
        }
        __syncthreads();
        // ---- vertical pass
        float* op = (ch == 0) ? o0 : (ch == 1) ? o1 : (ch == 2) ? o2 : o3;
#if USE_WMMA
        v8f acc = {};
        #pragma unroll
        for (int b = 0; b < 3; ++b) {
            #pragma unroll
            for (int t = 0; t < 4; ++t) {
                const int kk = 16 * b + 4 * t + 2 * half;   // K over H-rows 0..47
                // band: A[m][k] != 0 iff m <= k <= m+30, m = lane%16
                v2f av = { (kk     >= ln && kk     <= ln + 30) ? inv31 : 0.0f,
                           (kk + 1 >= ln && kk + 1 <= ln + 30) ? inv31 : 0.0f };
                v2f bv = { sH[kk * 17 + ln], sH[(kk + 1) * 17 + ln] };
                acc = __builtin_amdgcn_wmma_f32_16x16x4_f32(
                          false, av, false, bv, (short)0, acc, false, false);
            }
        }
        #pragma unroll
        for (int v = 0; v < 8; ++v)
            op[pb + (size_t)(ty + v + 8 * half) * PW + (tx + ln)] = acc[v];
#else
        for (int e = lane; e < 16 * 16; e += 32) {
            int m = e >> 4, n = e & 15;
            float s = 0.0f;
            for (int j = 0; j < 31; ++j) s += sH[(m + j) * 17 + n];
            op[pb + (size_t)(ty + m) * PW + (tx + n)] = s * inv31;
        }
#endif
        __syncthreads();
    }
}

// ============================================================================
// K3: guided-filter a/b   K5: q = box(a)*I + box(b), blend with x
// ============================================================================
__global__ __launch_bounds__(256)
void k_ab(const float* __restrict__ mI, const float* __restrict__ mp,
          const float* __restrict__ mIp, const float* __restrict__ mII,
          float* __restrict__ a, float* __restrict__ b)
{
    size_t i = (size_t)blockIdx.x * 256 + threadIdx.x;
    if (i >= (size_t)TOT) return;
    float vI = mI[i], vp = mp[i];
    float cov = mIp[i] - vI * vp;
    float var = mII[i] - vI * vI;
    float av = cov / (var + 1e-3f);
    a[i] = av;
    b[i] = vp - av * vI;
}

__global__ __launch_bounds__(256)
void k_blend(const float* __restrict__ ba, const float* __restrict__ bb,
             const float* __restrict__ gI, const float* __restrict__ x,
             const float* __restrict__ gwp, float* __restrict__ xg)
{
    size_t i = (size_t)blockIdx.x * 256 + threadIdx.x;
    if (i >= (size_t)TOT) return;
    float gw = gwp[0];
    float q = ba[i] * gI[i] + bb[i];
    xg[i] = gw * q + (1.0f - gw) * x[i];
}

// ============================================================================
// K6: adaptive contrast, 7x7 reflect-padded local mean/std + tanh remap
// ============================================================================
__global__ __launch_bounds__(256)
void k_contrast(const float* __restrict__ xg, float* __restrict__ xc)
{
    const int tx = blockIdx.x << 4, ty = blockIdx.y << 4, pl = blockIdx.z;
    const size_t pb = (size_t)pl * PLANE;
    const int tid = threadIdx.x, lx = tid & 15, ly = tid >> 4;

    __shared__ float sIn[22 * 23];
    __shared__ float sS [16 * 23];
    __shared__ float sQ [16 * 23];

    for (int i = tid; i < 22 * 22; i += 256) {
        int r = i / 22, c = i - r * 22;
        int gr = ty - 3 + r, gc = tx - 3 + c;
        gr = (gr < 0) ? -gr : ((gr >= PW) ? 2 * PW - 2 - gr : gr);   // reflect
        gc = (gc < 0) ? -gc : ((gc >= PW) ? 2 * PW - 2 - gc : gc);
        sIn[r * 23 + c] = xg[pb + (size_t)gr * PW + gc];
    }
    __syncthreads();
    for (int i = tid; i < 16 * 22; i += 256) {
        int r = i / 22, c = i - r * 22;
        float s = 0.0f, q = 0.0f;
        for (int k = 0; k < 7; ++k) { float v = sIn[(r + k) * 23 + c]; s += v; q += v * v; }
        sS[r * 23 + c] = s; sQ[r * 23 + c] = q;
    }
    __syncthreads();
    float s = 0.0f, q = 0.0f;
    for (int k = 0; k < 7; ++k) { s += sS[ly * 23 + lx + k]; q += sQ[ly * 23 + lx + k]; }
    const float inv49 = 1.0f / 49.0f;
    float m  = s * inv49;
    float vr = q * inv49 - m * m;
    float sd = sqrtf(fmaxf(vr, 0.0f) + 1e-8f);
    float v  = sIn[(ly + 3) * 23 + (lx + 3)];
    float ce = (v - m) / (sd + 1e-8f);
    xc[pb + (size_t)(ty + ly) * PW + (tx + lx)] = tanhf(ce) * sd + m;
}

// ============================================================================
// K7: edge conv (2->1 relu, 1->2 sigmoid, both 3x3 zero-pad) + final combine
// ============================================================================
__global__ __launch_bounds__(256)
void k_edgeconv(const float* __restrict__ xc, const float* __restrict__ ed,
                const float* __restrict__ w1, const float* __restrict__ b1,
                const float* __restrict__ w2, const float* __restrict__ b2,
                float* __restrict__ out)
{
    const int tx = blockIdx.x << 4, ty = blockIdx.y << 4, pl = blockIdx.z;
    const size_t pb = (size_t)pl * PLANE;
    const int tid = threadIdx.x, lx = tid & 15, ly = tid >> 4;

    __shared__ float sXc[20 * 21];
    __shared__ float sE [20 * 21];
    __shared__ float sH [18 * 19];

    for (int i = tid; i < 20 * 20; i += 256) {
        int r = i / 20, c = i - r * 20;
        int gr = ty - 2 + r, gc = tx - 2 + c;
        bool ok = ((unsigned)gr < (unsigned)PW) & ((unsigned)gc < (unsigned)PW);
        size_t gi = pb + (size_t)gr * PW + gc;
        sXc[r * 21 + c] = ok ? xc[gi] : 0.0f;
        sE [r * 21 + c] = ok ? ed[gi] : 0.0f;
    }
    float w1v[18], w2v[18];
    for (int j = 0; j < 18; ++j) { w1v[j] = w1[j]; w2v[j] = w2[j]; }
    float b1v = b1[0], b20 = b2[0], b21 = b2[1];
    __syncthreads();

    for (int i = tid; i < 18 * 18; i += 256) {
        int hr = i / 18, hc = i - hr * 18;
        float s = b1v;
        for (int ki = 0; ki < 3; ++ki)
            for (int kj = 0; kj < 3; ++kj) {
                int idx = (hr + ki) * 21 + (hc + kj);
                s += sXc[idx] * w1v[ki * 3 + kj] + sE[idx] * w1v[9 + ki * 3 + kj];
            }
        int gr = ty + hr - 1, gc = tx + hc - 1;
        bool ok = ((unsigned)gr < (unsigned)PW) & ((unsigned)gc < (unsigned)PW);
        sH[hr * 19 + hc] = ok ? fmaxf(s, 0.0f) : 0.0f;   // zero-pad h outside image
    }
    __syncthreads();

    float xcC = sXc[(ly + 2) * 21 + (lx + 2)];
    float edC = sE [(ly + 2) * 21 + (lx + 2)];
    for (int o = 0; o < 2; ++o) {
        float s = (o == 0) ? b20 : b21;
        for (int ki = 0; ki < 3; ++ki)
            for (int kj = 0; kj < 3; ++kj)
                s += sH[(ly + ki) * 19 + (lx + kj)] * w2v[o * 9 + ki * 3 + kj];
        float ref = 1.0f / (1.0f + expf(-s));
        out[((size_t)pl * 2 + o) * PLANE + (size_t)(ty + ly) * PW + (tx + lx)] =
            xcC + ref * edC;
    }
}

// ============================================================================
extern "C" void kernel_launch(void* const* d_in, const int* in_sizes, int n_in,
                              void* d_out, int out_size, void* d_ws, size_t ws_size,
                              hipStream_t stream)
{
    (void)in_sizes; (void)n_in; (void)out_size; (void)ws_size;

    const float* x    = (const float*)d_in[0];
    const float* guid = (const float*)d_in[1];
    const float* sw   = (const float*)d_in[2];
    /* d_in[3] = edge_strength: dead code in the reference */
    const float* gw   = (const float*)d_in[4];
    const float* w1   = (const float*)d_in[5];
    const float* b1   = (const float*)d_in[6];
    const float* w2   = (const float*)d_in[7];
    const float* b2   = (const float*)d_in[8];

    float* outf  = (float*)d_out;
    float* edges = outf + (size_t)2 * TOT;   // tuple output #2 lives in d_out tail

    // scratch: 6 full-stack f32 buffers = 96 MB (L2-resident on MI455X)
    float* ws = (float*)d_ws;
    float* m0 = ws;                         // mean(I)   -> later box(a)
    float* m1 = ws + 1 * (size_t)TOT;       // mean(p)   -> later box(b)
    float* m2 = ws + 2 * (size_t)TOT;       // mean(I*p)
    float* m3 = ws + 3 * (size_t)TOT;       // mean(I*I)
    float* ta = ws + 4 * (size_t)TOT;       // a,  then xg
    float* tb = ws + 5 * (size_t)TOT;       // b,  then xc

    dim3 tiles(PW / 16, PW / 16, NPL);      // 32 x 32 x 16 tile blocks
    int  n1d = (TOT + 255) / 256;

    k_edges   <<<tiles, 256, 0, stream>>>(x, sw, edges);
    k_box31   <<<tiles,  32, 0, stream>>>(guid, x, m0, m1, m2, m3, 4, 0);
    k_ab      <<<n1d,   256, 0, stream>>>(m0, m1, m2, m3, ta, tb);
    k_box31   <<<tiles,  32, 0, stream>>>(ta, tb, m0, m1, nullptr, nullptr, 2, 1);
    k_blend   <<<n1d,   256, 0, stream>>>(m0, m1, guid, x, gw, ta);
    k_contrast<<<tiles, 256, 0, stream>>>(ta, tb);
    k_edgeconv<<<tiles, 256, 0, stream>>>(tb, edges, w1, b1, w2, b2, outf);
}